// DisagreementRegularizer_987842478381
// MI455X (gfx1250) — compile-verified
//
#include <hip/hip_runtime.h>
#include <hip/hip_bf16.h>

// x: [B=128, Q=512, D=256] f32 ->  out[b] = -(1/Q^2) * || sum_q x[b,q,:]/max(||x[b,q,:]||,eps) ||^2
#define BATCH  128
#define QROWS  512
#define DDIM   256
#define NWAVES 16            // 512 threads / block
#define SLICES 4             // row-slices per batch (occupancy: 512 blocks, 8192 waves)
#define RPS    (QROWS / SLICES)
#define PPAD   (DDIM + 4)

typedef __attribute__((ext_vector_type(2))) float v2f;
typedef __attribute__((ext_vector_type(8))) float v8f;

// ---- wave32 all-lanes sum: 4x DPP16 ROW_XMASK adds + v_permlanex16 (no LDS traffic) ----
__device__ __forceinline__ float wave_reduce_sum(float v) {
#if __has_builtin(__builtin_amdgcn_update_dpp) && __has_builtin(__builtin_amdgcn_permlanex16)
    int t;
    t = __builtin_amdgcn_update_dpp(0, __builtin_bit_cast(int, v), 0x161, 0xf, 0xf, true); // xor 1
    v += __builtin_bit_cast(float, t);
    t = __builtin_amdgcn_update_dpp(0, __builtin_bit_cast(int, v), 0x162, 0xf, 0xf, true); // xor 2
    v += __builtin_bit_cast(float, t);
    t = __builtin_amdgcn_update_dpp(0, __builtin_bit_cast(int, v), 0x164, 0xf, 0xf, true); // xor 4
    v += __builtin_bit_cast(float, t);
    t = __builtin_amdgcn_update_dpp(0, __builtin_bit_cast(int, v), 0x168, 0xf, 0xf, true); // xor 8
    v += __builtin_bit_cast(float, t);
    int x = __builtin_bit_cast(int, v);
    t = __builtin_amdgcn_permlanex16(x, x, 0x76543210, 0xfedcba98, false, false);          // xor 16
    v += __builtin_bit_cast(float, t);
    return v;
#else
    #pragma unroll
    for (int off = 16; off > 0; off >>= 1) v += __shfl_xor(v, off, 32);
    return v;
#endif
}

// ================= Kernel 1: per (batch, slice) normalized row-sum partial =================
__global__ __launch_bounds__(512)
void partial_kernel(const float* __restrict__ x, float* __restrict__ ws) {
    __shared__ float P[NWAVES][PPAD];   // per-wave partial column sums

    const int b    = blockIdx.x / SLICES;
    const int s    = blockIdx.x % SLICES;
    const int tid  = threadIdx.x;
    const int lane = tid & 31;
    const int wave = tid >> 5;

    const float4* xb4  = (const float4*)(x + (size_t)b * QROWS * DDIM);
    const int     qend = (s + 1) * RPS;

    // ---- Phase 1: stream 128 rows, normalize, accumulate (lane owns 8 columns) ----
    float acc[8] = {0.f, 0.f, 0.f, 0.f, 0.f, 0.f, 0.f, 0.f};

    for (int q = s * RPS + wave; q < qend; q += NWAVES) {
        const float4* row = xb4 + (size_t)q * (DDIM / 4);
        float4 a = row[lane * 2 + 0];   // global_load_b128
        float4 c = row[lane * 2 + 1];   // global_load_b128

        if (q + NWAVES < qend)          // gfx1250 global_prefetch_b8
            __builtin_prefetch((const void*)(xb4 + (size_t)(q + NWAVES) * (DDIM / 4) + lane * 2), 0, 1);

        float ss = a.x * a.x + a.y * a.y + a.z * a.z + a.w * a.w
                 + c.x * c.x + c.y * c.y + c.z * c.z + c.w * c.w;
        ss = wave_reduce_sum(ss);

        float scale = 1.0f / fmaxf(sqrtf(ss), 1e-12f);

        acc[0] += a.x * scale;  acc[1] += a.y * scale;
        acc[2] += a.z * scale;  acc[3] += a.w * scale;
        acc[4] += c.x * scale;  acc[5] += c.y * scale;
        acc[6] += c.z * scale;  acc[7] += c.w * scale;
    }

    #pragma unroll
    for (int j = 0; j < 8; ++j)
        P[wave][lane * 8 + j] = acc[j];
    __syncthreads();

    // ---- Phase 2: fold 16 wave-partials -> 256-vector via chained WMMA (A = ones) ----
    // ws[(b*SLICES+s)*DDIM + :] = ones(1x16) @ P(16x256); wave w handles columns [16w,16w+16)
    {
        const int half = lane >> 4;
        const int n    = lane & 15;
        const int col  = wave * 16 + n;

        v8f csum = {};
        v2f ones = {1.0f, 1.0f};
        float fsum = 0.0f; (void)fsum;

        #pragma unroll
        for (int kk = 0; kk < 16; kk += 4) {
            v2f bb;
            bb[0] = P[kk + 2 * half + 0][col];
            bb[1] = P[kk + 2 * half + 1][col];
#if __has_builtin(__builtin_amdgcn_wmma_f32_16x16x4_f32)
            csum = __builtin_amdgcn_wmma_f32_16x16x4_f32(false, ones, false, bb,
                                                         (short)0, csum, false, false);
#else
            fsum += bb[0] + bb[1];
#endif
        }
#if __has_builtin(__builtin_amdgcn_wmma_f32_16x16x4_f32)
        float s_col = csum[0];                         // D[M=0][n] in lanes 0-15
#else
        float s_col = fsum + __shfl_xor(fsum, 16, 32);
#endif
        if (lane < 16)
            ws[(size_t)(b * SLICES + s) * DDIM + col] = s_col;
    }
}

// ================= Kernel 2: fold slices (K=4 -> one WMMA), ||S||^2, write out =================
__global__ __launch_bounds__(512)
void finalize_kernel(const float* __restrict__ ws, float* __restrict__ out) {
    __shared__ float R[NWAVES];

    const int b    = blockIdx.x;
    const int tid  = threadIdx.x;
    const int lane = tid & 31;
    const int wave = tid >> 5;

    const int half = lane >> 4;
    const int n    = lane & 15;
    const int col  = wave * 16 + n;

    const float* wb = ws + (size_t)b * SLICES * DDIM;

    // B = 4x16 tile: K = slice index, N = column within this wave's chunk
    v2f bb;
    bb[0] = wb[(size_t)(2 * half + 0) * DDIM + col];
    bb[1] = wb[(size_t)(2 * half + 1) * DDIM + col];

    v2f ones = {1.0f, 1.0f};
    v8f csum = {};
#if __has_builtin(__builtin_amdgcn_wmma_f32_16x16x4_f32)
    csum = __builtin_amdgcn_wmma_f32_16x16x4_f32(false, ones, false, bb,
                                                 (short)0, csum, false, false);
    float s_col = csum[0];                             // full 4-slice column sum
#else
    float fsum = bb[0] + bb[1];
    float s_col = fsum + __shfl_xor(fsum, 16, 32);
    (void)csum;
#endif

    // lanes 16-31 hold duplicate rows of D; count each column once
    float v = (lane < 16) ? s_col * s_col : 0.0f;
    v = wave_reduce_sum(v);
    if (lane == 0) R[wave] = v;
    __syncthreads();

    if (tid == 0) {
        float tot = 0.0f;
        #pragma unroll
        for (int w = 0; w < NWAVES; ++w) tot += R[w];
        out[b] = -tot / ((float)QROWS * (float)QROWS);
    }
}

extern "C" void kernel_launch(void* const* d_in, const int* in_sizes, int n_in,
                              void* d_out, int out_size, void* d_ws, size_t ws_size,
                              hipStream_t stream) {
    (void)in_sizes; (void)n_in; (void)ws_size;
    const float* x  = (const float*)d_in[0];
    float* out      = (float*)d_out;
    float* partials = (float*)d_ws;                     // BATCH*SLICES*DDIM floats = 512 KB

    const int nb = (out_size > 0) ? out_size : BATCH;
    partial_kernel <<<dim3(nb * SLICES), dim3(512), 0, stream>>>(x, partials);
    finalize_kernel<<<dim3(nb),          dim3(512), 0, stream>>>(partials, out);
}